// Model_59296318489279
// MI455X (gfx1250) — compile-verified
//
#include <hip/hip_runtime.h>
#include <hip/hip_bf16.h>

typedef __attribute__((ext_vector_type(2))) float v2f;
typedef __attribute__((ext_vector_type(8))) float v8f;

#define BB 2
#define VV 4
#define JJ 24
#define HMDIM 96
#define PLANE (HMDIM * HMDIM)
#define VOLD 48
#define NVOX (VOLD * VOLD * VOLD) /* 110592 */

// ---------------------------------------------------------------------------
// Kernel 1: project voxels (WMMA f32 16x16x4) + bilinear sample + view-sum.
// Wave layout: each wave owns 16 voxels. A = 16x4 matrix of projection rows
// (rows 0-5 = views 0,1; rows 8-13 = views 2,3; rest zero). B = 4x16 matrix of
// homogeneous voxel coords. D gives lane L (L<16): voxel L, views 0-1 in
// C[0..5]; lane L+16: voxel L, views 2-3 in C[0..5]. Each lane samples its
// two views for all 24 joints, then halves are folded with shfl_xor(16).
// ---------------------------------------------------------------------------
__global__ __launch_bounds__(256) void vol_sample_kernel(
    const float* __restrict__ hm, const float* __restrict__ proj,
    const float* __restrict__ coord, float* __restrict__ vol) {
  const int lane = threadIdx.x & 31;
  const int wave = (blockIdx.x << 3) + (threadIdx.x >> 5);
  const int b = wave / (NVOX / 16);
  const int group = wave - b * (NVOX / 16);
  const int m = lane & 15;
  const bool hi = lane >= 16;
  const int voxel = (group << 4) + m;

  // --- A operand: rearranged projection rows, lane half selects K pair -----
  float a0 = 0.0f, a1 = 0.0f;
  {
    int view = -1, row = 0;
    if (m < 6) {
      view = m / 3;
      row = m - 3 * view;
    } else if (m >= 8 && m < 14) {
      int t = m - 8;
      view = 2 + t / 3;
      row = t - 3 * (t / 3);
    }
    if (view >= 0) {
      const int k0 = hi ? 2 : 0;
      const float* p = proj + (((b * VV + view) * 3) + row) * 4;
      a0 = p[k0];
      a1 = p[k0 + 1];
    }
  }

  // --- B operand: homogeneous voxel coordinates (branchless: no exec-mask
  // divergence in front of the WMMA) ---------------------------------------
  float b0, b1;
  {
    const float* c = coord + ((size_t)(b * NVOX + voxel)) * 3;
    b0 = c[hi ? 2 : 0];           // K=0 row: x  | K=2 row: z
    const float cy = c[1];        // always loaded
    b1 = hi ? 1.0f : cy;          // K=1 row: y  | K=3 row: homogeneous 1
  }

  v2f A;
  A.x = a0;
  A.y = a1;
  v2f Bv;
  Bv.x = b0;
  Bv.y = b1;
  v8f C = {};
  // D = A(16x4) * B(4x16) + 0 : exact f32 projection of 16 voxels x 4 views
  C = __builtin_amdgcn_wmma_f32_16x16x4_f32(false, A, false, Bv, (short)0, C,
                                            false, false);

  float acc[JJ];
#pragma unroll
  for (int j = 0; j < JJ; ++j) acc[j] = 0.0f;

  const int vbase = hi ? 2 : 0;
#pragma unroll
  for (int vv = 0; vv < 2; ++vv) {
    const float X = C[vv * 3 + 0];
    const float Y = C[vv * 3 + 1];
    const float Z = C[vv * 3 + 2];
    const float zs = (Z == 0.0f) ? 1.0f : Z;
    const float vvalid = (Z > 0.0f) ? 1.0f : 0.0f;
    // Perspective divide: v_rcp_f32 (~1 ulp) is plenty for a sampling coord.
    const float inv = __builtin_amdgcn_rcpf(zs);
    // ix = px*(HM-1)/HM  (gx = 2*(px/HM - .5); ix = (gx+1)*.5*(HM-1))
    const float scale = 95.0f / 96.0f;
    float ix = (X * inv) * scale;
    float iy = (Y * inv) * scale;
    ix = fminf(fmaxf(ix, -8.0f), 104.0f);
    iy = fminf(fmaxf(iy, -8.0f), 104.0f);
    const float x0f = floorf(ix), y0f = floorf(iy);
    const float wx1 = ix - x0f, wy1 = iy - y0f;
    const float wx0 = 1.0f - wx1, wy0 = 1.0f - wy1;
    const int x0 = (int)x0f, y0 = (int)y0f;
    const int x1 = x0 + 1, y1 = y0 + 1;
    const float vx0 = (x0 >= 0 && x0 < HMDIM) ? 1.0f : 0.0f;
    const float vx1 = (x1 >= 0 && x1 < HMDIM) ? 1.0f : 0.0f;
    const float vy0 = (y0 >= 0 && y0 < HMDIM) ? 1.0f : 0.0f;
    const float vy1 = (y1 >= 0 && y1 < HMDIM) ? 1.0f : 0.0f;
    const float w00 = wy0 * wx0 * vy0 * vx0 * vvalid;
    const float w01 = wy0 * wx1 * vy0 * vx1 * vvalid;
    const float w10 = wy1 * wx0 * vy1 * vx0 * vvalid;
    const float w11 = wy1 * wx1 * vy1 * vx1 * vvalid;
    const int cx0 = min(max(x0, 0), HMDIM - 1);
    const int cx1 = min(max(x1, 0), HMDIM - 1);
    const int cy0 = min(max(y0, 0), HMDIM - 1);
    const int cy1 = min(max(y1, 0), HMDIM - 1);
    const int o00 = cy0 * HMDIM + cx0, o01 = cy0 * HMDIM + cx1;
    const int o10 = cy1 * HMDIM + cx0, o11 = cy1 * HMDIM + cx1;
    const float* base =
        hm + ((size_t)((b * VV + (vbase + vv)) * JJ)) * PLANE;
    __builtin_prefetch(base + o00, 0, 0);  // global_prefetch_b8 into L2
#pragma unroll
    for (int j = 0; j < JJ; ++j) {
      const float* pj = base + j * PLANE;
      acc[j] += w00 * pj[o00] + w01 * pj[o01] + w10 * pj[o10] + w11 * pj[o11];
    }
  }

  // fold view-pairs: lanes L and L+16 hold the same voxel, different views
#pragma unroll
  for (int j = 0; j < JJ; ++j) acc[j] += __shfl_xor(acc[j], 16, 32);

  if (!hi) {
#pragma unroll
    for (int j = 0; j < JJ; ++j)
      vol[((size_t)(b * JJ + j)) * NVOX + voxel] = acc[j];
  }
}

// ---------------------------------------------------------------------------
// Kernel 2: per-(b,j) softmax over 110592 voxels + coordinate expectation.
// One workgroup per (b,j) row (48 blocks).
// ---------------------------------------------------------------------------
__device__ inline float block_reduce_sum(float v, float* s) {
  const int tid = threadIdx.x;
  s[tid] = v;
  __syncthreads();
  for (int st = 128; st > 0; st >>= 1) {
    if (tid < st) s[tid] += s[tid + st];
    __syncthreads();
  }
  const float r = s[0];
  __syncthreads();
  return r;
}

__global__ __launch_bounds__(256) void softmax_coord_kernel(
    const float* __restrict__ vol, const float* __restrict__ coord,
    float* __restrict__ out) {
  const int bj = blockIdx.x;
  const int b = bj / JJ;
  const float* v = vol + (size_t)bj * NVOX;
  const float* c = coord + (size_t)b * NVOX * 3;
  __shared__ float sred[256];
  const int tid = threadIdx.x;

  float mloc = -3.402823e38f;
  for (int i = tid; i < NVOX; i += 256) mloc = fmaxf(mloc, v[i]);
  sred[tid] = mloc;
  __syncthreads();
  for (int st = 128; st > 0; st >>= 1) {
    if (tid < st) sred[tid] = fmaxf(sred[tid], sred[tid + st]);
    __syncthreads();
  }
  const float mx = sred[0];
  __syncthreads();

  float se = 0.0f, s0 = 0.0f, s1 = 0.0f, s2 = 0.0f;
  for (int i = tid; i < NVOX; i += 256) {
    const float e = __expf(v[i] - mx);
    se += e;
    s0 += e * c[i * 3 + 0];
    s1 += e * c[i * 3 + 1];
    s2 += e * c[i * 3 + 2];
  }
  se = block_reduce_sum(se, sred);
  s0 = block_reduce_sum(s0, sred);
  s1 = block_reduce_sum(s1, sred);
  s2 = block_reduce_sum(s2, sred);
  if (tid == 0) {
    const float inv = 1.0f / se;
    out[bj * 3 + 0] = s0 * inv;
    out[bj * 3 + 1] = s1 * inv;
    out[bj * 3 + 2] = s2 * inv;
  }
}

extern "C" void kernel_launch(void* const* d_in, const int* in_sizes, int n_in,
                              void* d_out, int out_size, void* d_ws,
                              size_t ws_size, hipStream_t stream) {
  const float* hm = (const float*)d_in[0];     // (2,4,24,96,96)
  const float* proj = (const float*)d_in[1];   // (2,4,3,4)
  const float* coord = (const float*)d_in[2];  // (2,48,48,48,3)
  float* out = (float*)d_out;                  // (2,24,3)
  float* vol = (float*)d_ws;                   // (2,24,110592) f32 scratch

  const int waves = BB * (NVOX / 16);       // 13824 waves, 16 voxels each
  dim3 g1(waves / 8), blk(256);             // 8 waves per 256-thread block
  vol_sample_kernel<<<g1, blk, 0, stream>>>(hm, proj, coord, vol);

  dim3 g2(BB * JJ);
  softmax_coord_kernel<<<g2, blk, 0, stream>>>(vol, coord, out);
}